// Mask_Attention_23390391894461
// MI455X (gfx1250) — compile-verified
//
#include <hip/hip_runtime.h>
#include <math.h>

// Mask-attention (flash style) for MI455X / gfx1250, wave32.
// f32 WMMA 16x16x4 for both matmuls; async global->LDS double buffering.
// B=8, NQ=2048, NC=2048, D=128. Compute-bound (AI ~680 FLOP/byte).

typedef __attribute__((ext_vector_type(2))) float v2f;
typedef __attribute__((ext_vector_type(8))) float v8f;
typedef __attribute__((ext_vector_type(4))) int   v4i;

#define B_       8
#define NQ_      2048
#define NC_      2048
#define D_       128
#define TILE_C   64          // context rows per block (16 per wave)
#define TILE_Q   32          // query rows per LDS tile
#define NWAVES   4
#define NTHREADS 128
#define QS_STRIDE 132        // padded dword stride for Q tile rows (bank-conflict free)
#define PS_STRIDE 36         // padded dword stride for P scratch rows

#if defined(__has_builtin)
#  if __has_builtin(__builtin_amdgcn_global_load_async_to_lds_b128)
#    define HAVE_ASYNC_LDS 1
#  endif
#endif
#ifndef HAVE_ASYNC_LDS
#  define HAVE_ASYNC_LDS 0
#endif

#define GAS __attribute__((address_space(1)))   // global
#define LAS __attribute__((address_space(3)))   // LDS

// Stage one 32x128 f32 Q tile into LDS (padded rows). Async path uses
// GLOBAL_LOAD_ASYNC_TO_LDS_B128 (ASYNCcnt), no VGPR round trip.
__device__ __forceinline__ void stage_q_tile(const float* gsrc, float* ldst, int tid)
{
#if HAVE_ASYNC_LDS
    #pragma unroll
    for (int j = 0; j < 8; ++j) {
        const int idx  = tid + NTHREADS * j;   // 0..1023 float4s of the tile
        const int row  = idx >> 5;
        const int col4 = idx & 31;
        __builtin_amdgcn_global_load_async_to_lds_b128(
            (GAS v4i*)(gsrc + (size_t)row * D_ + col4 * 4),
            (LAS v4i*)(ldst + row * QS_STRIDE + col4 * 4), 0, 0);
    }
#else
    #pragma unroll
    for (int j = 0; j < 8; ++j) {
        const int idx  = tid + NTHREADS * j;
        const int row  = idx >> 5;
        const int col4 = idx & 31;
        float4 v = *reinterpret_cast<const float4*>(gsrc + (size_t)row * D_ + col4 * 4);
        *reinterpret_cast<float4*>(ldst + row * QS_STRIDE + col4 * 4) = v;
    }
#endif
}

__device__ __forceinline__ void wait_async_done()
{
#if HAVE_ASYNC_LDS
#  if defined(__has_builtin) && __has_builtin(__builtin_amdgcn_s_wait_asynccnt)
    __builtin_amdgcn_s_wait_asynccnt(0);
#  else
    asm volatile("s_wait_asynccnt 0" ::: "memory");
#  endif
#endif
    // Fallback path stages through VGPRs/DS; __syncthreads' dscnt waits cover it.
}

__global__ __launch_bounds__(NTHREADS)
void mask_attn_kernel(const float* __restrict__ query,
                      const float* __restrict__ context,
                      float* __restrict__ out)
{
    // Two distinct shared objects (not one [2][...] array) so alias analysis can
    // prove async writes to the staging buffer don't touch the compute buffer.
    __shared__ float QsA[TILE_Q * QS_STRIDE];
    __shared__ float QsB[TILE_Q * QS_STRIDE];
    __shared__ float Ps[NWAVES][16 * PS_STRIDE];   // per-wave P transpose scratch

    const int tid  = threadIdx.x;
    const int lane = tid & 31;
    const int wid  = tid >> 5;
    const int n    = lane & 15;   // N index within 16-lane group
    const int hf   = lane >> 4;   // 0 = lanes 0-15, 1 = lanes 16-31
    const int b    = blockIdx.y;
    const int c0   = blockIdx.x * TILE_C;
    const int crow = c0 + wid * 16 + n;            // A-matrix row (M = lane%16)

    // ---- Preload this wave's context A-fragments: 16 rows x K=128, VGPR-resident.
    // A layout (32-bit, 16x4): VGPR0 = K {0 | 2}, VGPR1 = K {1 | 3} by lane half.
    v2f aC[32];
    {
        const float* cp = context + ((size_t)b * NC_ + crow) * D_ + 2 * hf;
        #pragma unroll
        for (int k = 0; k < 32; ++k)
            aC[k] = *reinterpret_cast<const v2f*>(cp + 4 * k);
    }

    // ---- Flash-attention state: O (16x128), running row max m, running row sum l.
    v8f O[8];
    #pragma unroll
    for (int t = 0; t < 8; ++t) O[t] = v8f{};
    v8f m, l;
    #pragma unroll
    for (int r = 0; r < 8; ++r) { m[r] = -INFINITY; l[r] = 0.0f; }

    const float* qbase = query + (size_t)b * NQ_ * D_;
    float* Pw = Ps[wid];

    // One q-tile step: S = C@Qt^T (64 WMMA), masked online softmax, O += P@Qt (64 WMMA).
    auto compute_tile = [&](const float* __restrict__ Qcur) {
        // ---- S strip 16x32, K=128. B layout (4x16): VGPR0 = K{0|2}, VGPR1 = K{1|3}.
        v8f s0 = v8f{}, s1 = v8f{};
        #pragma unroll
        for (int k = 0; k < 32; ++k) {
            const int d0 = 4 * k + 2 * hf;
            v2f a  = aC[k];
            v2f b0 = *reinterpret_cast<const v2f*>(&Qcur[n * QS_STRIDE + d0]);
            v2f b1 = *reinterpret_cast<const v2f*>(&Qcur[(n + 16) * QS_STRIDE + d0]);
            s0 = __builtin_amdgcn_wmma_f32_16x16x4_f32(false, a, false, b0, (short)0, s0, false, false);
            s1 = __builtin_amdgcn_wmma_f32_16x16x4_f32(false, a, false, b1, (short)0, s1, false, false);
        }

        // ---- Mask (score==0 -> -inf) + online softmax update.
        v8f rowmax;
        #pragma unroll
        for (int r = 0; r < 8; ++r) {
            float a0 = s0[r]; if (a0 == 0.0f) a0 = -INFINITY;
            float a1 = s1[r]; if (a1 == 0.0f) a1 = -INFINITY;
            s0[r] = a0; s1[r] = a1;
            rowmax[r] = fmaxf(a0, a1);
        }
        #pragma unroll
        for (int off = 8; off >= 1; off >>= 1)
            #pragma unroll
            for (int r = 0; r < 8; ++r)
                rowmax[r] = fmaxf(rowmax[r], __shfl_xor(rowmax[r], off, 16));

        v8f scale, rowsum;
        #pragma unroll
        for (int r = 0; r < 8; ++r) {
            float mn = fmaxf(m[r], rowmax[r]);
            scale[r] = (m[r] == mn) ? 1.0f : __expf(m[r] - mn);  // safe for -inf==-inf
            m[r] = mn;
            float p0 = (s0[r] == -INFINITY) ? 0.0f : __expf(s0[r] - mn);
            float p1 = (s1[r] == -INFINITY) ? 0.0f : __expf(s1[r] - mn);
            s0[r] = p0; s1[r] = p1;
            rowsum[r] = p0 + p1;
        }
        #pragma unroll
        for (int off = 8; off >= 1; off >>= 1)
            #pragma unroll
            for (int r = 0; r < 8; ++r)
                rowsum[r] += __shfl_xor(rowsum[r], off, 16);
        #pragma unroll
        for (int r = 0; r < 8; ++r) l[r] = l[r] * scale[r] + rowsum[r];

        // Rescale O: per-row scale shares the replicated C/D lane layout -> pure VALU.
        #pragma unroll
        for (int t = 0; t < 8; ++t)
            #pragma unroll
            for (int r = 0; r < 8; ++r)
                O[t][r] *= scale[r];

        // ---- C-layout -> A-layout for P via per-wave padded LDS scratch.
        #pragma unroll
        for (int r = 0; r < 8; ++r) {
            Pw[(r + 8 * hf) * PS_STRIDE + n]      = s0[r];
            Pw[(r + 8 * hf) * PS_STRIDE + n + 16] = s1[r];
        }
        v2f aP[8];
        #pragma unroll
        for (int k = 0; k < 8; ++k)
            aP[k] = *reinterpret_cast<const v2f*>(&Pw[n * PS_STRIDE + 4 * k + 2 * hf]);

        // ---- O += P @ Q_tile : K=32 over q, 8 D-column tiles (64 WMMAs).
        #pragma unroll
        for (int t = 0; t < 8; ++t) {
            v8f acc = O[t];
            #pragma unroll
            for (int k = 0; k < 8; ++k) {
                const int k0 = 4 * k + 2 * hf;
                v2f bq;
                bq.x = Qcur[(k0    ) * QS_STRIDE + 16 * t + n];
                bq.y = Qcur[(k0 + 1) * QS_STRIDE + 16 * t + n];
                acc = __builtin_amdgcn_wmma_f32_16x16x4_f32(false, aP[k], false, bq, (short)0, acc, false, false);
            }
            O[t] = acc;
        }
    };

    // ---- Software pipeline, unrolled x2 with compile-time buffer selection.
    // Hazard proof: the buffer staged at step i was last read at step i-1, and
    // every wave passed wait_async+barrier in between.
    stage_q_tile(qbase, QsA, tid);
    for (int qi = 0; qi < NQ_ / TILE_Q; qi += 2) {
        wait_async_done();
        __syncthreads();
        stage_q_tile(qbase + (size_t)(qi + 1) * TILE_Q * D_, QsB, tid);  // qi+1 <= 63 always
        compute_tile(QsA);

        wait_async_done();
        __syncthreads();
        if (qi + 2 < NQ_ / TILE_Q)
            stage_q_tile(qbase + (size_t)(qi + 2) * TILE_Q * D_, QsA, tid);
        compute_tile(QsB);
    }

    // ---- Epilogue: normalize (l==0 reproduces the NaN->0 patch) and store.
    v8f inv;
    #pragma unroll
    for (int r = 0; r < 8; ++r) inv[r] = (l[r] > 0.0f) ? (1.0f / l[r]) : 0.0f;

    float* obase = out + ((size_t)b * NC_ + c0 + wid * 16 + 8 * hf) * (size_t)D_;
    #pragma unroll
    for (int t = 0; t < 8; ++t)
        #pragma unroll
        for (int r = 0; r < 8; ++r)
            obase[(size_t)r * D_ + 16 * t + n] = O[t][r] * inv[r];
}

extern "C" void kernel_launch(void* const* d_in, const int* in_sizes, int n_in,
                              void* d_out, int out_size, void* d_ws, size_t ws_size,
                              hipStream_t stream) {
    (void)in_sizes; (void)n_in; (void)out_size; (void)d_ws; (void)ws_size;
    const float* query   = (const float*)d_in[0];
    const float* context = (const float*)d_in[1];
    float*       out     = (float*)d_out;
    dim3 grid(NC_ / TILE_C, B_);   // 32 x 8 = 256 blocks
    mask_attn_kernel<<<grid, NTHREADS, 0, stream>>>(query, context, out);
}